// TFDWConvTranspose2d_807453851750
// MI455X (gfx1250) — compile-verified
//
#include <hip/hip_runtime.h>

// Depthwise ConvTranspose2d k=4, s=2, p=1 on (16,64,64,256) -> (16,128,128,256), f32.
// Memory-bound: ~335 MB HBM traffic -> ~14us floor at 23.3 TB/s.
// CDNA5 paths used: GLOBAL_LOAD_ASYNC_TO_LDS_B128 staging (ASYNCcnt),
// B128 vector accesses, non-temporal output stores (keep L2-resident input hot).

#define B_   16
#define H_   64
#define W_   64
#define C_   256
#define OH_  128
#define OW_  128
#define C4_  64     // C/4 float4 groups

typedef float v4f __attribute__((ext_vector_type(4)));
typedef int   v4i __attribute__((vector_size(16)));   // matches builtin param type
typedef __attribute__((address_space(1))) v4i* as1_v4i;
typedef __attribute__((address_space(3))) v4i* as3_v4i;

#if defined(__has_builtin)
#  if __has_builtin(__builtin_amdgcn_global_load_async_to_lds_b128)
#    define HAVE_ASYNC_B128 1
#  endif
#  if __has_builtin(__builtin_amdgcn_s_wait_asynccnt)
#    define HAVE_WAIT_ASYNC 1
#  endif
#endif

__device__ __forceinline__ void async_ld_b128(const v4f* gsrc, v4f* ldst) {
#ifdef HAVE_ASYNC_B128
  v4i* g = (v4i*)gsrc;                 // drop const, reinterpret pointee
  v4i* l = (v4i*)ldst;
  __builtin_amdgcn_global_load_async_to_lds_b128(
      (as1_v4i)g, (as3_v4i)l, 0, 0);   // (global src, lds dst, imm off, imm cpol)
#else
  unsigned lds_off =
      (unsigned)(unsigned long long)(__attribute__((address_space(3))) void*)ldst;
  asm volatile("global_load_async_to_lds_b128 %0, %1, off"
               :: "v"(lds_off), "v"((unsigned long long)gsrc) : "memory");
#endif
}

__device__ __forceinline__ void wait_async0() {
#ifdef HAVE_WAIT_ASYNC
  __builtin_amdgcn_s_wait_asynccnt(0);
#else
  asm volatile("s_wait_asynccnt 0" ::: "memory");
#endif
}

__global__ __launch_bounds__(256) void TFDWConvTranspose2d_kernel(
    const float* __restrict__ x,   // (16,64,64,256)
    const float* __restrict__ w,   // (4,4,1,256)
    const float* __restrict__ bias,// (256,)
    float* __restrict__ out)       // (16,128,128,256)
{
  __shared__ v4f lds[2][4][C4_];   // 2 input rows x 4 input cols x 256 ch = 8 KB

  const int tid = threadIdx.x;          // 0..255
  const int ow0 = blockIdx.x * 4;       // output-column tile base
  const int oh  = (int)blockIdx.y;
  const int b   = (int)blockIdx.z;

  const int ihA    = (oh + 1) >> 1;     // row paired with tap khA
  const int ihB    = ihA - 1;           // row paired with tap khA+2
  const int khA    = (oh + 1) & 1;
  const int iwBase = (ow0 >> 1) - 1;    // leftmost staged input column

  // ---- Stage 2x4 input pixels (all 256 ch) into LDS: 512 B128 slots, 2/thread.
  const v4f* x4 = (const v4f*)x;
#pragma unroll
  for (int i = 0; i < 2; ++i) {
    const int slot = tid + i * 256;
    const int c4   = slot & 63;
    const int pix  = slot >> 6;         // 0..7
    const int r    = pix >> 2;          // 0..1  (0 -> ihB, 1 -> ihA)
    const int l    = pix & 3;           // 0..3  (iw = iwBase + l)
    const int ih   = ihB + r;
    const int iw   = iwBase + l;
    v4f* ldst = &lds[r][l][c4];
    if ((unsigned)ih < (unsigned)H_ && (unsigned)iw < (unsigned)W_) {
      const v4f* gsrc = x4 + (((b * H_ + ih) * W_ + iw) * C4_ + c4);
      async_ld_b128(gsrc, ldst);        // ASYNCcnt path, 16B per lane
    } else {
      v4f z = {0.f, 0.f, 0.f, 0.f};     // zero-fill halo (DS store, masked lanes)
      *ldst = z;
    }
  }
  wait_async0();
  __syncthreads();

  // ---- Compute: one (ow, 4-channel) output per thread.
  const int owi  = tid >> 6;            // 0..3
  const int c4   = tid & 63;
  const int ow   = ow0 + owi;
  const int kwA  = (ow + 1) & 1;
  const int iwA  = (ow + 1) >> 1;
  const int colA = iwA - iwBase;        // 1..3
  const int colB = colA - 1;            // 0..2

  const v4f* w4 = (const v4f*)w;        // w[(kh*4+kw)*64 + c4]
  const v4f wAA = w4[(khA * 4 + kwA) * C4_ + c4];
  const v4f wAB = w4[(khA * 4 + kwA + 2) * C4_ + c4];
  const v4f wBA = w4[((khA + 2) * 4 + kwA) * C4_ + c4];
  const v4f wBB = w4[((khA + 2) * 4 + kwA + 2) * C4_ + c4];
  const v4f bb  = ((const v4f*)bias)[c4];

  const v4f xAA = lds[1][colA][c4];     // (ihA, iwA) -> tap (khA,   kwA)
  const v4f xAB = lds[1][colB][c4];     // (ihA, iwB) -> tap (khA,   kwA+2)
  const v4f xBA = lds[0][colA][c4];     // (ihB, iwA) -> tap (khA+2, kwA)
  const v4f xBB = lds[0][colB][c4];     // (ihB, iwB) -> tap (khA+2, kwA+2)

  v4f acc = bb;
  acc += xAA * wAA;
  acc += xAB * wAB;
  acc += xBA * wBA;
  acc += xBB * wBB;

  v4f* o = (v4f*)out + (((b * OH_ + oh) * OW_ + ow) * C4_ + c4);
  __builtin_nontemporal_store(acc, o);  // streaming 268MB: don't pollute L2
}

extern "C" void kernel_launch(void* const* d_in, const int* in_sizes, int n_in,
                              void* d_out, int out_size, void* d_ws, size_t ws_size,
                              hipStream_t stream) {
  (void)in_sizes; (void)n_in; (void)out_size; (void)d_ws; (void)ws_size;
  const float* x    = (const float*)d_in[0];
  const float* w    = (const float*)d_in[1];
  const float* bias = (const float*)d_in[2];
  float* out        = (float*)d_out;

  dim3 grid(OW_ / 4, OH_, B_);   // (32, 128, 16) blocks
  dim3 block(256);
  hipLaunchKernelGGL(TFDWConvTranspose2d_kernel, grid, block, 0, stream,
                     x, w, bias, out);
}